// TriangleProjection_27358941676285
// MI455X (gfx1250) — compile-verified
//
#include <hip/hip_runtime.h>
#include <hip/hip_bf16.h>

// ---------------------------------------------------------------------------
// TriangleProjection: project y_pred (B x 2, f32) onto triangles (B x 6, f32).
// Memory-bound: 80 MB total traffic, ~3.4 us floor at 23.3 TB/s HBM (fits in
// the 192 MB L2 across graph replays). Strategy: async global->LDS b128
// staging of the 24 B/elem AoS triangle data (CDNA5
// GLOBAL_LOAD_ASYNC_TO_LDS_B128 + s_wait_asynccnt), float4 path for
// y_pred / output, 2 elements per lane, wave32 blocks of 256.
// ---------------------------------------------------------------------------

#ifndef HAVE_ASYNC_LDS
#if defined(__has_builtin)
#if __has_builtin(__builtin_amdgcn_global_load_async_to_lds_b128)
#define HAVE_ASYNC_LDS 1
#endif
#endif
#endif
#ifndef HAVE_ASYNC_LDS
#define HAVE_ASYNC_LDS 0
#endif

#ifndef HAVE_WAIT_ASYNC
#if defined(__has_builtin)
#if __has_builtin(__builtin_amdgcn_s_wait_asynccnt)
#define HAVE_WAIT_ASYNC 1
#endif
#endif
#endif
#ifndef HAVE_WAIT_ASYNC
#define HAVE_WAIT_ASYNC 0
#endif

typedef int v4i __attribute__((ext_vector_type(4)));
typedef __attribute__((address_space(1))) char as1_char;
typedef __attribute__((address_space(3))) char as3_char;
typedef __attribute__((address_space(1))) v4i as1_v4i;
typedef __attribute__((address_space(3))) v4i as3_v4i;

#define TP_THREADS 256
#define TP_EPB 512 // elements per block (2 per thread)

__device__ __forceinline__ float tp_safe_div(float num, float den, bool c) {
    // matches reference: where(c, num, 0) / where(c, den, 1)
    return (c ? num : 0.0f) / (c ? den : 1.0f);
}

__device__ __forceinline__ float2 tp_project(float px, float py,
                                             float ax, float ay,
                                             float bx, float by,
                                             float cx, float cy) {
    float abx = bx - ax, aby = by - ay;
    float acx = cx - ax, acy = cy - ay;
    float apx = px - ax, apy = py - ay;
    float d1 = abx * apx + aby * apy;
    float d2 = acx * apx + acy * apy;
    float bpx = px - bx, bpy = py - by;
    float d3 = abx * bpx + aby * bpy;
    float d4 = acx * bpx + acy * bpy;
    float cpx = px - cx, cpy = py - cy;
    float d5 = abx * cpx + aby * cpy;
    float d6 = acx * cpx + acy * cpy;

    float vc = d1 * d4 - d3 * d2;
    float vb = d5 * d2 - d1 * d6;
    float va = d3 * d6 - d5 * d4;

    bool cond_a  = (d1 <= 0.0f) && (d2 <= 0.0f);
    bool cond_b  = (d3 >= 0.0f) && (d4 <= d3);
    bool cond_ab = (vc <= 0.0f) && (d1 >= 0.0f) && (d3 <= 0.0f);
    bool cond_c  = (d6 >= 0.0f) && (d5 <= d6);
    bool cond_ac = (vb <= 0.0f) && (d2 >= 0.0f) && (d6 <= 0.0f);
    bool cond_bc = (va <= 0.0f) && ((d4 - d3) >= 0.0f) && ((d5 - d6) >= 0.0f);

    float t_ab = tp_safe_div(d1, d1 - d3, cond_ab);
    float t_ac = tp_safe_div(d2, d2 - d6, cond_ac);
    float t_bc = tp_safe_div(d4 - d3, (d4 - d3) + (d5 - d6), cond_bc);

    float denom = va + vb + vc;
    float inv = 1.0f / ((fabsf(denom) > 0.0f) ? denom : 1.0f);
    float v = vb * inv;
    float w = vc * inv;

    float rx = ax + v * abx + w * acx;
    float ry = ay + v * aby + w * acy;
    // precedence matches reference where-stack (later override wins, a last)
    if (cond_bc) { rx = bx + t_bc * (cx - bx); ry = by + t_bc * (cy - by); }
    if (cond_ac) { rx = ax + t_ac * acx;       ry = ay + t_ac * acy; }
    if (cond_c)  { rx = cx;                    ry = cy; }
    if (cond_ab) { rx = ax + t_ab * abx;       ry = ay + t_ab * aby; }
    if (cond_b)  { rx = bx;                    ry = by; }
    if (cond_a)  { rx = ax;                    ry = ay; }
    return make_float2(rx, ry);
}

__global__ __launch_bounds__(TP_THREADS) void
TriangleProjection_27358941676285_kernel(const float* __restrict__ y_pred,
                                         const float* __restrict__ tri,
                                         float* __restrict__ out, int B) {
    const int tid  = threadIdx.x;
    const int base = blockIdx.x * TP_EPB;

    if (base + TP_EPB <= B) {
        // ---------------- fast path: full block of 512 elements -------------
        float T[12]; // two triangle records for this lane

#if HAVE_ASYNC_LDS
        // Stage this block's 12288 B of AoS triangle data into LDS using the
        // CDNA5 async data path: 3 rounds x (256 lanes x b128) of perfectly
        // contiguous global reads, tracked by ASYNCcnt.
        __shared__ float lds_tri[6 * TP_EPB]; // 12288 bytes
        {
            as1_char* g = (as1_char*)(tri + (size_t)base * 6) + tid * 16;
            as3_char* l = (as3_char*)lds_tri + tid * 16;
            __builtin_amdgcn_global_load_async_to_lds_b128(
                (as1_v4i*)g, (as3_v4i*)l, 0, 0);
            __builtin_amdgcn_global_load_async_to_lds_b128(
                (as1_v4i*)(g + 4096), (as3_v4i*)(l + 4096), 0, 0);
            __builtin_amdgcn_global_load_async_to_lds_b128(
                (as1_v4i*)(g + 8192), (as3_v4i*)(l + 8192), 0, 0);
        }
#if HAVE_WAIT_ASYNC
        __builtin_amdgcn_s_wait_asynccnt(0);
#else
        asm volatile("s_wait_asynccnt 0" ::: "memory");
#endif
        __syncthreads();
        {
            const float* lt = &lds_tri[tid * 12];
#pragma unroll
            for (int i = 0; i < 12; ++i) T[i] = lt[i];
        }
#else
        // Fallback: direct b128 loads (48 B/lane, wave covers 1536 B/instr).
        {
            const float* g = tri + (size_t)base * 6 + tid * 12;
            float4 a0 = *(const float4*)(g);
            float4 a1 = *(const float4*)(g + 4);
            float4 a2 = *(const float4*)(g + 8);
            T[0] = a0.x; T[1] = a0.y; T[2]  = a0.z; T[3]  = a0.w;
            T[4] = a1.x; T[5] = a1.y; T[6]  = a1.z; T[7]  = a1.w;
            T[8] = a2.x; T[9] = a2.y; T[10] = a2.z; T[11] = a2.w;
        }
#endif
        // y_pred for 2 elements: one aligned b128 load
        const float4 p2 = ((const float4*)y_pred)[(size_t)(base >> 1) + tid];

        float2 r0 = tp_project(p2.x, p2.y, T[0], T[1], T[2], T[3], T[4], T[5]);
        float2 r1 = tp_project(p2.z, p2.w, T[6], T[7], T[8], T[9], T[10], T[11]);

        ((float4*)out)[(size_t)(base >> 1) + tid] =
            make_float4(r0.x, r0.y, r1.x, r1.y);
    } else {
        // ---------------- tail path: guarded scalar elements ----------------
#pragma unroll
        for (int k = 0; k < 2; ++k) {
            int idx = base + tid * 2 + k;
            if (idx < B) {
                const float* g = tri + (size_t)idx * 6;
                float2 p = ((const float2*)y_pred)[idx];
                float2 r = tp_project(p.x, p.y, g[0], g[1], g[2], g[3], g[4], g[5]);
                ((float2*)out)[idx] = r;
            }
        }
    }
}

extern "C" void kernel_launch(void* const* d_in, const int* in_sizes, int n_in,
                              void* d_out, int out_size, void* d_ws, size_t ws_size,
                              hipStream_t stream) {
    (void)n_in; (void)d_ws; (void)ws_size; (void)out_size;
    const float* y_pred = (const float*)d_in[0];       // B x 2 f32
    const float* tri    = (const float*)d_in[1];       // B x 6 f32
    float* out          = (float*)d_out;               // B x 2 f32
    const int B = in_sizes[0] / 2;
    const int blocks = (B + TP_EPB - 1) / TP_EPB;
    TriangleProjection_27358941676285_kernel<<<blocks, TP_THREADS, 0, stream>>>(
        y_pred, tri, out, B);
}